// QCNN_Shared_2c_12756052869703
// MI455X (gfx1250) — compile-verified
//
#include <hip/hip_runtime.h>
#include <math.h>

// MI455X / gfx1250, wave32. One sample per wave: 8-qubit state = 256 complex
// amps = 8 complex per lane (16 VGPRs per psi). Qubits 0..4 = lane bits
// (shuffle butterflies), qubits 5..7 = register bits (pure VALU).
// Final 4->15->1 MLP done with V_WMMA_F32_16X16X4_F32 (16 samples / block).

typedef __attribute__((ext_vector_type(2))) float v2f;
typedef __attribute__((ext_vector_type(8))) float v8f;

#define DEV static __device__ __forceinline__

DEV float shx(float v, int m) { return __shfl_xor(v, m, 32); }

struct CMat2 { float r00, i00, r01, i01, r10, i10, r11, i11; };

// Generic complex 2x2 one-qubit gate.
DEV void ap1c(float re[8], float im[8], CMat2 u, int q, int lane) {
  if (q >= 5) {
    const int tb = 1 << (7 - q);
#pragma unroll
    for (int r = 0; r < 8; ++r) {
      if ((r & tb) == 0) {
        const int r1 = r | tb;
        float a0r = re[r], a0i = im[r], a1r = re[r1], a1i = im[r1];
        re[r]  = u.r00 * a0r - u.i00 * a0i + u.r01 * a1r - u.i01 * a1i;
        im[r]  = u.r00 * a0i + u.i00 * a0r + u.r01 * a1i + u.i01 * a1r;
        re[r1] = u.r10 * a0r - u.i10 * a0i + u.r11 * a1r - u.i11 * a1i;
        im[r1] = u.r10 * a0i + u.i10 * a0r + u.r11 * a1i + u.i11 * a1r;
      }
    }
  } else {
    const int lm = 1 << (4 - q);
    const bool b = (lane & lm) != 0;
    const float dr = b ? u.r11 : u.r00, di = b ? u.i11 : u.i00;
    const float gr = b ? u.r10 : u.r01, gi = b ? u.i10 : u.i01;
#pragma unroll
    for (int r = 0; r < 8; ++r) {
      float sr = re[r], si = im[r];
      float tr = shx(sr, lm), ti = shx(si, lm);
      re[r] = dr * sr - di * si + gr * tr - gi * ti;
      im[r] = dr * si + di * sr + gr * ti + gi * tr;
    }
  }
}

// Real 2x2 one-qubit gate (RY etc).
DEV void ap1r(float re[8], float im[8], float m00, float m01, float m10,
              float m11, int q, int lane) {
  if (q >= 5) {
    const int tb = 1 << (7 - q);
#pragma unroll
    for (int r = 0; r < 8; ++r) {
      if ((r & tb) == 0) {
        const int r1 = r | tb;
        float a0r = re[r], a1r = re[r1], a0i = im[r], a1i = im[r1];
        re[r]  = m00 * a0r + m01 * a1r;
        re[r1] = m10 * a0r + m11 * a1r;
        im[r]  = m00 * a0i + m01 * a1i;
        im[r1] = m10 * a0i + m11 * a1i;
      }
    }
  } else {
    const int lm = 1 << (4 - q);
    const bool b = (lane & lm) != 0;
    const float d = b ? m11 : m00;
    const float o = b ? m10 : m01;
#pragma unroll
    for (int r = 0; r < 8; ++r) {
      float orr = shx(re[r], lm), oii = shx(im[r], lm);
      re[r] = d * re[r] + o * orr;
      im[r] = d * im[r] + o * oii;
    }
  }
}

// RZ(t): diag(e^{-it/2}, e^{+it/2}); c=cos(t/2), s=sin(t/2).
DEV void rzg(float re[8], float im[8], float c, float s, int q, int lane) {
  if (q >= 5) {
    const int tb = 1 << (7 - q);
#pragma unroll
    for (int r = 0; r < 8; ++r) {
      float sb = (r & tb) ? s : -s;
      float nr = c * re[r] - sb * im[r];
      float ni = c * im[r] + sb * re[r];
      re[r] = nr; im[r] = ni;
    }
  } else {
    const int lm = 1 << (4 - q);
    const float sb = (lane & lm) ? s : -s;
#pragma unroll
    for (int r = 0; r < 8; ++r) {
      float nr = c * re[r] - sb * im[r];
      float ni = c * im[r] + sb * re[r];
      re[r] = nr; im[r] = ni;
    }
  }
}

// CNOT with control qc, target qt.
DEV void cxg(float re[8], float im[8], int qc, int qt, int lane) {
  if (qt >= 5) {
    const int tb = 1 << (7 - qt);
    if (qc >= 5) {
      const int cb = 1 << (7 - qc);
#pragma unroll
      for (int r = 0; r < 8; ++r) {
        if (((r & tb) == 0) && (r & cb)) {
          const int r1 = r | tb;
          float t;
          t = re[r]; re[r] = re[r1]; re[r1] = t;
          t = im[r]; im[r] = im[r1]; im[r1] = t;
        }
      }
    } else {
      const int lm = 1 << (4 - qc);
      const bool cbit = (lane & lm) != 0;
#pragma unroll
      for (int r = 0; r < 8; ++r) {
        if ((r & tb) == 0) {
          const int r1 = r | tb;
          float n0r = cbit ? re[r1] : re[r];
          float n1r = cbit ? re[r] : re[r1];
          float n0i = cbit ? im[r1] : im[r];
          float n1i = cbit ? im[r] : im[r1];
          re[r] = n0r; re[r1] = n1r; im[r] = n0i; im[r1] = n1i;
        }
      }
    }
  } else {
    const int tm = 1 << (4 - qt);
    if (qc >= 5) {
      const int cb = 1 << (7 - qc);
#pragma unroll
      for (int r = 0; r < 8; ++r) {
        if (r & cb) {
          re[r] = shx(re[r], tm);
          im[r] = shx(im[r], tm);
        }
      }
    } else {
      const int lm = 1 << (4 - qc);
      const bool cbit = (lane & lm) != 0;
#pragma unroll
      for (int r = 0; r < 8; ++r) {
        float orr = shx(re[r], tm), oii = shx(im[r], tm);
        re[r] = cbit ? orr : re[r];
        im[r] = cbit ? oii : im[r];
      }
    }
  }
}

// CRX: on control=1 subspace, RX(t): re' = c*re + s*im_other; im' = c*im - s*re_other.
DEV void crxg(float re[8], float im[8], float cs, float sn, int qc, int qt,
              int lane) {
  if (qt >= 5) {
    const int tb = 1 << (7 - qt);
    if (qc >= 5) {
      const int cb = 1 << (7 - qc);
#pragma unroll
      for (int r = 0; r < 8; ++r) {
        if (((r & tb) == 0) && (r & cb)) {
          const int r1 = r | tb;
          float n0r = cs * re[r] + sn * im[r1];
          float n0i = cs * im[r] - sn * re[r1];
          float n1r = cs * re[r1] + sn * im[r];
          float n1i = cs * im[r1] - sn * re[r];
          re[r] = n0r; im[r] = n0i; re[r1] = n1r; im[r1] = n1i;
        }
      }
    } else {
      const int lm = 1 << (4 - qc);
      const bool cbit = (lane & lm) != 0;
#pragma unroll
      for (int r = 0; r < 8; ++r) {
        if ((r & tb) == 0) {
          const int r1 = r | tb;
          float n0r = cs * re[r] + sn * im[r1];
          float n0i = cs * im[r] - sn * re[r1];
          float n1r = cs * re[r1] + sn * im[r];
          float n1i = cs * im[r1] - sn * re[r];
          re[r]  = cbit ? n0r : re[r];
          im[r]  = cbit ? n0i : im[r];
          re[r1] = cbit ? n1r : re[r1];
          im[r1] = cbit ? n1i : im[r1];
        }
      }
    }
  } else {
    const int tm = 1 << (4 - qt);
    if (qc >= 5) {
      const int cb = 1 << (7 - qc);
#pragma unroll
      for (int r = 0; r < 8; ++r) {
        if (r & cb) {
          float orr = shx(re[r], tm), oii = shx(im[r], tm);
          float nr = cs * re[r] + sn * oii;
          float ni = cs * im[r] - sn * orr;
          re[r] = nr; im[r] = ni;
        }
      }
    } else {
      const int lm = 1 << (4 - qc);
      const bool cbit = (lane & lm) != 0;
#pragma unroll
      for (int r = 0; r < 8; ++r) {
        float orr = shx(re[r], tm), oii = shx(im[r], tm);
        float nr = cs * re[r] + sn * oii;
        float ni = cs * im[r] - sn * orr;
        re[r] = cbit ? nr : re[r];
        im[r] = cbit ? ni : im[r];
      }
    }
  }
}

DEV CMat2 mk_u3(float t, float p, float l) {
  float ct, st, cp, sp, cl, sl, cpl, spl;
  sincosf(0.5f * t, &st, &ct);
  sincosf(p, &sp, &cp);
  sincosf(l, &sl, &cl);
  sincosf(p + l, &spl, &cpl);
  CMat2 u;
  u.r00 = ct;        u.i00 = 0.f;
  u.r01 = -cl * st;  u.i01 = -sl * st;
  u.r10 = cp * st;   u.i10 = sp * st;
  u.r11 = cpl * ct;  u.i11 = spl * ct;
  return u;
}

__global__ __launch_bounds__(512) void qcnn_wave_kernel(
    const float* __restrict__ x, const float* __restrict__ rz_p,
    const float* __restrict__ ry_p, const float* __restrict__ ry2_p,
    const float* __restrict__ crx_p, const float* __restrict__ u3_p,
    const float* __restrict__ u3b_p, const float* __restrict__ W1,
    const float* __restrict__ b1, const float* __restrict__ W2,
    const float* __restrict__ b2, float* __restrict__ out, int B) {
  __shared__ float feats[16 * 4];

  const int lane = threadIdx.x & 31;
  const int wave = threadIdx.x >> 5;
  const int sample = blockIdx.x * 16 + wave;

  // ----- state init: |0...0> -----
  float pr[8], pi[8], qr[8], qi[8];
#pragma unroll
  for (int r = 0; r < 8; ++r) { pr[r] = 0.f; pi[r] = 0.f; }
  pr[0] = (lane == 0) ? 1.f : 0.f;

  // ----- angle encoding: 4 cycles of RY + CNOT ring -----
  float xv = (sample < B) ? x[sample * 32 + lane] : 0.f;
  float se, ce;
  sincosf(0.5f * xv, &se, &ce);
#pragma unroll
  for (int c = 0; c < 4; ++c) {
#pragma unroll
    for (int q = 0; q < 8; ++q) {
      float cq = __shfl(ce, c * 8 + q, 32);
      float sq = __shfl(se, c * 8 + q, 32);
      ap1r(pr, pi, cq, -sq, sq, cq, q, lane);
    }
#pragma unroll
    for (int q = 0; q < 8; ++q) cxg(pr, pi, q, (q + 1) & 7, lane);
  }

  // psi1 = copy of encoded state
#pragma unroll
  for (int r = 0; r < 8; ++r) { qr[r] = pr[r]; qi[r] = pi[r]; }

  const float RT = 0.70710678118654752f;

  // ----- two layers -----
#pragma unroll
  for (int layer = 0; layer < 2; ++layer) {
    const int n = (layer == 0) ? 8 : 4;
    float cz, sz, cy, sy, cy2, sy2, ccx, scx;
    sincosf(0.5f * rz_p[layer], &sz, &cz);
    sincosf(0.5f * ry_p[layer], &sy, &cy);
    sincosf(0.5f * ry2_p[layer], &sy2, &cy2);
    sincosf(0.5f * crx_p[layer], &scx, &ccx);
    CMat2 U3a = mk_u3(u3_p[layer * 3 + 0], u3_p[layer * 3 + 1], u3_p[layer * 3 + 2]);
    CMat2 U3b = mk_u3(u3b_p[layer * 3 + 0], u3b_p[layer * 3 + 1], u3b_p[layer * 3 + 2]);

    // conv blocks on psi0
#pragma unroll
    for (int parity = 0; parity < 2; ++parity) {
#pragma unroll
      for (int ii = 0; ii < 4; ++ii) {
        const int i = parity + 2 * ii;
        if (i < n - 1) {
          const int qc = (layer == 0) ? i : (2 * i + 1);
          const int qt = (layer == 0) ? (i + 1) : (2 * i + 3);
          rzg(pr, pi, RT, -RT, qt, lane);                 // RZ(-pi/2) on qt
          cxg(pr, pi, qt, qc, lane);                      // CX(qt -> qc)
          rzg(pr, pi, cz, sz, qc, lane);                  // RZ(rz_p) on qc
          ap1r(pr, pi, cy, -sy, sy, cy, qt, lane);        // RY(ry_p) on qt
          cxg(pr, pi, qc, qt, lane);                      // CX(qc -> qt)
          ap1r(pr, pi, cy2, -sy2, sy2, cy2, qt, lane);    // RY(ry2_p) on qt
          cxg(pr, pi, qt, qc, lane);                      // CX(qt -> qc)
          rzg(pr, pi, RT, RT, qc, lane);                  // RZ(+pi/2) on qc
        }
      }
    }
    // U3a on remaining qubits of psi0: layer0 -> {1,3,5,7}, layer1 -> {3,7}
#pragma unroll
    for (int k = 0; k < 4; ++k) {
      const int q = (layer == 0) ? (2 * k + 1) : ((k < 2) ? (4 * k + 3) : -1);
      if (q >= 0) ap1c(pr, pi, U3a, q, lane);
    }
    // CRX ladder on psi1
#pragma unroll
    for (int parity = 0; parity < 2; ++parity) {
#pragma unroll
      for (int ii = 0; ii < 4; ++ii) {
        const int i = parity + 2 * ii;
        if (i < n - 1) {
          const int qc = (layer == 0) ? i : (2 * i + 1);
          const int qt = (layer == 0) ? (i + 1) : (2 * i + 3);
          crxg(qr, qi, ccx, scx, qc, qt, lane);
        }
      }
    }
    // U3b on remaining qubits of psi1
#pragma unroll
    for (int k = 0; k < 4; ++k) {
      const int q = (layer == 0) ? (2 * k + 1) : ((k < 2) ? (4 * k + 3) : -1);
      if (q >= 0) ap1c(qr, qi, U3b, q, lane);
    }
  }

  // ----- features: <Z3>, <Z7> for psi0 and psi1 -----
  // qubit 3 -> lane bit 1; qubit 7 -> register bit 0
  float z3p = 0.f, z7p = 0.f, z3q = 0.f, z7q = 0.f;
#pragma unroll
  for (int r = 0; r < 8; ++r) {
    float p0 = pr[r] * pr[r] + pi[r] * pi[r];
    float p1 = qr[r] * qr[r] + qi[r] * qi[r];
    float sg7 = (r & 1) ? -1.f : 1.f;
    z3p += p0;       z7p += sg7 * p0;
    z3q += p1;       z7q += sg7 * p1;
  }
  const float sg3 = ((lane >> 1) & 1) ? -1.f : 1.f;
  z3p *= sg3; z3q *= sg3;
#pragma unroll
  for (int m = 1; m < 32; m <<= 1) {
    z3p += shx(z3p, m);
    z7p += shx(z7p, m);
    z3q += shx(z3q, m);
    z7q += shx(z7q, m);
  }
  if (lane == 0) {
    feats[wave * 4 + 0] = z3p;
    feats[wave * 4 + 1] = z7p;
    feats[wave * 4 + 2] = z3q;
    feats[wave * 4 + 3] = z7q;
  }
  __syncthreads();

  // ----- MLP via V_WMMA_F32_16X16X4_F32 on wave 0 -----
  // A (16x4): lane l holds M=l&15, K = (l>>4)*2 + {0,1}
  // B (4x16): VGPR0/1 hold K rows {0,1} in lanes 0-15, {2,3} in lanes 16-31
  // C/D (16x16): VGPR j = row M=j (lanes 0-15) / M=j+8 (lanes 16-31), N=l&15
  if (threadIdx.x < 32) {
    const int l = threadIdx.x;
    const int nn = l & 15;
    const int half = l >> 4;
    v2f A, Bv;
    A.x = feats[nn * 4 + half * 2 + 0];
    A.y = feats[nn * 4 + half * 2 + 1];
    Bv.x = (nn < 15) ? W1[nn * 4 + half * 2 + 0] : 0.f;  // B[k][n] = W1[n][k]
    Bv.y = (nn < 15) ? W1[nn * 4 + half * 2 + 1] : 0.f;
    const float bias = (nn < 15) ? b1[nn] : 0.f;
    v8f C;
#pragma unroll
    for (int j = 0; j < 8; ++j) C[j] = bias;
    v8f D = __builtin_amdgcn_wmma_f32_16x16x4_f32(false, A, false, Bv,
                                                  (short)0, C, false, false);
    const float w2n = (nn < 15) ? W2[nn] : 0.f;
    const float bb = b2[0];
#pragma unroll
    for (int j = 0; j < 8; ++j) {
      float h = tanhf(D[j]);
      float t = h * w2n;
      t += shx(t, 1); t += shx(t, 2); t += shx(t, 4); t += shx(t, 8);
      float o = 1.f / (1.f + expf(-(t + bb)));
      if (nn == 0) {
        int idx = blockIdx.x * 16 + half * 8 + j;
        if (idx < B) out[idx] = o;
      }
    }
  }
}

extern "C" void kernel_launch(void* const* d_in, const int* in_sizes, int n_in,
                              void* d_out, int out_size, void* d_ws,
                              size_t ws_size, hipStream_t stream) {
  (void)n_in; (void)out_size; (void)d_ws; (void)ws_size;
  const float* x     = (const float*)d_in[0];
  const float* rz_p  = (const float*)d_in[1];
  const float* ry_p  = (const float*)d_in[2];
  const float* ry2_p = (const float*)d_in[3];
  const float* crx_p = (const float*)d_in[4];
  const float* u3_p  = (const float*)d_in[5];
  const float* u3b_p = (const float*)d_in[6];
  const float* W1    = (const float*)d_in[7];
  const float* b1    = (const float*)d_in[8];
  const float* W2    = (const float*)d_in[9];
  const float* b2    = (const float*)d_in[10];
  float* out = (float*)d_out;

  const int B = in_sizes[0] / 32;           // x is (B, 32)
  const int blocks = (B + 15) / 16;         // 16 samples (waves) per block
  qcnn_wave_kernel<<<blocks, 512, 0, stream>>>(x, rz_p, ry_p, ry2_p, crx_p,
                                               u3_p, u3b_p, W1, b1, W2, b2,
                                               out, B);
}